// QuantumLayer_28819230556445
// MI455X (gfx1250) — compile-verified
//
#include <hip/hip_runtime.h>

// MI455X / gfx1250 quantum-circuit kernel, WMMA-fused gate version.
//
// - One batch element per workgroup; full 128KB complex64 state resident in
//   LDS (320KB/WGP on CDNA5). HBM traffic ~60KB total -> compute/LDS bound.
// - Commuting single-qubit gates fused into 16x16 tensor-product unitaries,
//   applied as complex GEMMs via V_WMMA_F32_16X16X4_F32 (4 real 16x16x16
//   products = 16 chained K=4 WMMAs per 16-column tile).
// - CNOT ring = predicated LDS swaps (pure permutation, no flops).
// - Final z @ W.T + b + x done with the same f32 WMMA shape.

#define NQ    14
#define DIM   16384        // 2^14 amplitudes
#define BATCH 512
#define TPB   256          // 8 wave32 waves per workgroup
#define PER   (DIM / TPB)  // 64 amplitudes per thread
#define NTILE (DIM / 256)  // 64 column tiles of 16 per fused sweep
#define NWAVE (TPB / 32)   // 8

typedef float v2f __attribute__((ext_vector_type(2)));
typedef float v8f __attribute__((ext_vector_type(8)));

struct C22 { float ur[2][2]; float ui[2][2]; };

__device__ __forceinline__ C22 ry_mat(float theta) {
    C22 m;
    float c = __cosf(0.5f * theta), s = __sinf(0.5f * theta);
    m.ur[0][0] = c;   m.ur[0][1] = -s;  m.ur[1][0] = s;   m.ur[1][1] = c;
    m.ui[0][0] = 0.f; m.ui[0][1] = 0.f; m.ui[1][0] = 0.f; m.ui[1][1] = 0.f;
    return m;
}

// PennyLane Rot(phi,theta,omega) = RZ(omega) RY(theta) RZ(phi)
__device__ __forceinline__ C22 rot_mat(float phi, float th, float om) {
    C22 m;
    float ct = __cosf(0.5f * th), st = __sinf(0.5f * th);
    float A = 0.5f * (phi + om), Bb = 0.5f * (phi - om);
    float cA = __cosf(A), sA = __sinf(A), cB = __cosf(Bb), sB = __sinf(Bb);
    m.ur[0][0] =  ct * cA;  m.ui[0][0] = -ct * sA;   // e^{-iA} ct
    m.ur[0][1] = -st * cB;  m.ui[0][1] = -st * sB;   // -e^{+iB} st
    m.ur[1][0] =  st * cB;  m.ui[1][0] = -st * sB;   // e^{-iB} st
    m.ur[1][1] =  ct * cA;  m.ui[1][1] =  ct * sA;   // e^{+iA} ct
    return m;
}

__device__ __forceinline__ v8f wmma4(v2f a, v2f b, v8f c) {
    return __builtin_amdgcn_wmma_f32_16x16x4_f32(
        false, a, false, b, (short)0, c, false, false);
}

// G = m[0] (x) m[1] (x) m[2] (x) m[3]  (m[0] on the block's MSB qubit).
// 256 threads: thread tid computes entry (row a = tid>>4, col c = tid&15).
__device__ __forceinline__ void build_G(float* Gr, float* Gi, const C22 m[4], int tid) {
    const int a = tid >> 4, c = tid & 15;
    float gr = 1.f, gi = 0.f;
#pragma unroll
    for (int t = 0; t < 4; ++t) {
        int ab = (a >> (3 - t)) & 1;
        int bb = (c >> (3 - t)) & 1;
        float ur = ab ? (bb ? m[t].ur[1][1] : m[t].ur[1][0])
                      : (bb ? m[t].ur[0][1] : m[t].ur[0][0]);
        float ui = ab ? (bb ? m[t].ui[1][1] : m[t].ui[1][0])
                      : (bb ? m[t].ui[0][1] : m[t].ui[0][0]);
        float nr = gr * ur - gi * ui;
        gi = gr * ui + gi * ur;
        gr = nr;
    }
    Gr[tid] = gr;
    Gi[tid] = gi;
}

// Apply 16x16 complex G to state on 4 adjacent qubits whose bits occupy
// positions s+3..s. State viewed as 16 x 1024 column matrix; column j's
// amplitude for row a sits at idx = ((j>>s)<<(s+4)) | (a<<s) | (j & (2^s-1)).
// Each wave owns disjoint 16-column tiles -> in-place, no intra-sweep barrier.
// Dr = Gr*Sr - Gi*Si ; Di = Gr*Si + Gi*Sr  (f32 WMMA has no A-negate, so
// the -Gi term uses a pre-negated copy of Gi).
__device__ __forceinline__ void wmma_apply16(float* re, float* im,
                                             const float* Gr, const float* Gi,
                                             int s, int tid) {
    const int lane = tid & 31, wave = tid >> 5;
    const int hi = lane >> 4, ln = lane & 15;
    const int Rm1 = (1 << s) - 1;

    // Preload A (G) chunks: A 16x4 layout -> lane (hi,ln) holds
    // G[ln][4c+2hi], G[ln][4c+2hi+1] in v0,v1.
    v2f Ar[4], Ai[4], An[4];
#pragma unroll
    for (int c = 0; c < 4; ++c) {
        int k = 4 * c + 2 * hi;
        Ar[c].x = Gr[ln * 16 + k];  Ar[c].y = Gr[ln * 16 + k + 1];
        Ai[c].x = Gi[ln * 16 + k];  Ai[c].y = Gi[ln * 16 + k + 1];
        An[c].x = -Ai[c].x;         An[c].y = -Ai[c].y;
    }

    for (int t = wave; t < NTILE; t += NWAVE) {
        int j = t * 16 + ln;                                  // this lane's column
        int colbase = ((j >> s) << (s + 4)) | (j & Rm1);
        // B 4x16 layout: lane (hi,ln) holds rows 4c+2hi, 4c+2hi+1 at col ln.
        v2f Br[4], Bi[4];
#pragma unroll
        for (int c = 0; c < 4; ++c) {
            int k = 4 * c + 2 * hi;
            Br[c].x = re[colbase + (k << s)];
            Br[c].y = re[colbase + ((k + 1) << s)];
            Bi[c].x = im[colbase + (k << s)];
            Bi[c].y = im[colbase + ((k + 1) << s)];
        }
        v8f Dr = {}, Di = {};
#pragma unroll
        for (int c = 0; c < 4; ++c) {
            Dr = wmma4(Ar[c], Br[c], Dr);
            Di = wmma4(Ar[c], Bi[c], Di);
            Di = wmma4(Ai[c], Br[c], Di);
            Dr = wmma4(An[c], Bi[c], Dr);
        }
        // D layout: vgpr v, lane (hi,ln) -> row v+8hi, col ln.
#pragma unroll
        for (int v = 0; v < 8; ++v) {
            int off = colbase + ((v + 8 * hi) << s);
            re[off] = Dr[v];
            im[off] = Di[v];
        }
    }
}

// 4x4 complex gate on the two LSB qubits (bits 1,0): groups of 4 consecutive
// amplitudes, scalar per-thread.
__device__ __forceinline__ void apply4x4(float* re, float* im,
                                         const C22 mA, const C22 mB, int tid) {
    float g4r[4][4], g4i[4][4];
#pragma unroll
    for (int a = 0; a < 4; ++a)
#pragma unroll
        for (int c = 0; c < 4; ++c) {
            float xr = mA.ur[a >> 1][c >> 1], xi = mA.ui[a >> 1][c >> 1];
            float yr = mB.ur[a & 1][c & 1],   yi = mB.ui[a & 1][c & 1];
            g4r[a][c] = xr * yr - xi * yi;
            g4i[a][c] = xr * yi + xi * yr;
        }
    for (int k = 0; k < DIM / 4 / TPB; ++k) {
        int base = (k * TPB + tid) << 2;
        float rr[4], ii[4];
#pragma unroll
        for (int c = 0; c < 4; ++c) { rr[c] = re[base + c]; ii[c] = im[base + c]; }
#pragma unroll
        for (int a = 0; a < 4; ++a) {
            float nr = 0.f, ni = 0.f;
#pragma unroll
            for (int c = 0; c < 4; ++c) {
                nr += g4r[a][c] * rr[c] - g4i[a][c] * ii[c];
                ni += g4r[a][c] * ii[c] + g4i[a][c] * rr[c];
            }
            re[base + a] = nr;
            im[base + a] = ni;
        }
    }
}

// ---------------------------------------------------------------------------
// Kernel 1: full simulation, one workgroup per batch element.
// ---------------------------------------------------------------------------
__global__ __launch_bounds__(TPB) void circuit_kernel(
    const float* __restrict__ x,       // (BATCH, NQ)
    const float* __restrict__ w,       // (2, NQ, 3)
    float* __restrict__ zout)          // (BATCH, 16) padded workspace
{
    extern __shared__ float smem[];
    float* re   = smem;                // [DIM]
    float* im   = smem + DIM;          // [DIM]
    float* Gr   = smem + 2 * DIM;      // [256]
    float* Gi   = Gr + 256;            // [256]
    float* zred = Gi + 256;            // [16]

    const int b   = blockIdx.x;
    const int tid = threadIdx.x;

    // init |0...0>
    for (int k = 0; k < PER; ++k) {
        int i = k * TPB + tid;
        re[i] = (i == 0) ? 1.0f : 0.0f;
        im[i] = 0.0f;
    }
    __syncthreads();

    // ---- AngleEmbedding: RY gates, fused 4 qubits at a time ----
    for (int blk = 0; blk < 3; ++blk) {
        C22 m[4];
#pragma unroll
        for (int t = 0; t < 4; ++t) m[t] = ry_mat(x[b * NQ + blk * 4 + t]);
        build_G(Gr, Gi, m, tid);
        __syncthreads();
        wmma_apply16(re, im, Gr, Gi, 10 - 4 * blk, tid);
        __syncthreads();
    }
    apply4x4(re, im, ry_mat(x[b * NQ + 12]), ry_mat(x[b * NQ + 13]), tid);
    __syncthreads();

    // ---- StronglyEntanglingLayers ----
    for (int l = 0; l < 2; ++l) {
        for (int blk = 0; blk < 3; ++blk) {
            C22 m[4];
#pragma unroll
            for (int t = 0; t < 4; ++t) {
                const float* p = w + (l * NQ + blk * 4 + t) * 3;
                m[t] = rot_mat(p[0], p[1], p[2]);
            }
            build_G(Gr, Gi, m, tid);
            __syncthreads();
            wmma_apply16(re, im, Gr, Gi, 10 - 4 * blk, tid);
            __syncthreads();
        }
        {
            const float* pA = w + (l * NQ + 12) * 3;
            const float* pB = w + (l * NQ + 13) * 3;
            apply4x4(re, im, rot_mat(pA[0], pA[1], pA[2]),
                             rot_mat(pB[0], pB[1], pB[2]), tid);
            __syncthreads();
        }
        // CNOT ring (reference order matters: consecutive CNOTs don't commute)
        int r = (l % (NQ - 1)) + 1;
        for (int q = 0; q < NQ; ++q) {
            int t  = (q + r) % NQ;
            int mc = 1 << (NQ - 1 - q);
            int mt = 1 << (NQ - 1 - t);
            for (int k = 0; k < PER; ++k) {
                int i = k * TPB + tid;
                if ((i & mc) && !(i & mt)) {
                    int j = i | mt;
                    float tr = re[i]; re[i] = re[j]; re[j] = tr;
                    float ti = im[i]; im[i] = im[j]; im[j] = ti;
                }
            }
            __syncthreads();
        }
    }

    // ---- <Z_q> ----
    float zacc[NQ];
#pragma unroll
    for (int q = 0; q < NQ; ++q) zacc[q] = 0.0f;
    for (int k = 0; k < PER; ++k) {
        int i = k * TPB + tid;
        float pr = re[i] * re[i] + im[i] * im[i];
#pragma unroll
        for (int q = 0; q < NQ; ++q)
            zacc[q] += ((i >> (NQ - 1 - q)) & 1) ? -pr : pr;
    }
    __syncthreads();
    if (tid < 16) zred[tid] = 0.0f;
    __syncthreads();
    const int lane = tid & 31;
#pragma unroll
    for (int q = 0; q < NQ; ++q) {
        float v = zacc[q];
#pragma unroll
        for (int off = 16; off > 0; off >>= 1) v += __shfl_xor(v, off, 32);
        if (lane == 0) atomicAdd(&zred[q], v);   // ds_add_f32
    }
    __syncthreads();
    if (tid < 16) zout[b * 16 + tid] = zred[tid];  // cols 14,15 are zero-pad
}

// ---------------------------------------------------------------------------
// Kernel 2: out = Z @ W.T + bias + x via V_WMMA_F32_16X16X4_F32.
// Z comes K-padded (512x16), so A loads are unconditional.
// ---------------------------------------------------------------------------
__global__ __launch_bounds__(32) void out_gemm(
    const float* __restrict__ z,     // (BATCH, 16) padded
    const float* __restrict__ W,     // (NQ, NQ)
    const float* __restrict__ bias,  // (NQ)
    const float* __restrict__ x,     // (BATCH, NQ)
    float* __restrict__ out)         // (BATCH, NQ)
{
    const int rowbase = blockIdx.x * 16;
    const int lane = threadIdx.x;
    const int hi = lane >> 4, ln = lane & 15;

    v8f acc = {};
#pragma unroll
    for (int k0 = 0; k0 < 16; k0 += 4) {
        int ka = k0 + 2 * hi;
        v2f a, bb;
        a.x = z[(rowbase + ln) * 16 + ka];
        a.y = z[(rowbase + ln) * 16 + ka + 1];
        bb.x = (ka     < NQ && ln < NQ) ? W[ln * NQ + ka]     : 0.0f;
        bb.y = (ka + 1 < NQ && ln < NQ) ? W[ln * NQ + ka + 1] : 0.0f;
        acc = wmma4(a, bb, acc);
    }
#pragma unroll
    for (int j = 0; j < 8; ++j) {
        int m = rowbase + j + hi * 8;
        int n = ln;
        if (n < NQ)
            out[m * NQ + n] = acc[j] + bias[n] + x[m * NQ + n];
    }
}

extern "C" void kernel_launch(void* const* d_in, const int* in_sizes, int n_in,
                              void* d_out, int out_size, void* d_ws, size_t ws_size,
                              hipStream_t stream) {
    const float* x    = (const float*)d_in[0];   // (512,14) f32
    const float* wgt  = (const float*)d_in[1];   // (2,14,3) f32
    const float* W    = (const float*)d_in[2];   // (14,14)  f32
    const float* bias = (const float*)d_in[3];   // (14)     f32
    float* out  = (float*)d_out;                 // (512,14) f32
    float* zbuf = (float*)d_ws;                  // (512,16) f32 scratch

    size_t lds_bytes = (size_t)(2 * DIM + 256 + 256 + 16) * sizeof(float); // 133184 B
    circuit_kernel<<<BATCH, TPB, lds_bytes, stream>>>(x, wgt, zbuf);
    out_gemm<<<BATCH / 16, 32, 0, stream>>>(zbuf, W, bias, x, out);
}